// GCNBlockTimeInv_5600637354463
// MI455X (gfx1250) — compile-verified
//
#include <hip/hip_runtime.h>
#include <stdint.h>

typedef float v2f __attribute__((ext_vector_type(2)));
typedef float v8f __attribute__((ext_vector_type(8)));

#define FEAT 128

// Native f32 global atomic add, no return (STOREcnt), device scope.
__device__ __forceinline__ void gatomic_fadd(float* p, float v) {
  uint64_t a = (uint64_t)(uintptr_t)p;
  asm volatile("global_atomic_add_f32 %0, %1, off scope:SCOPE_DEV"
               :
               : "v"(a), "v"(v)
               : "memory");
}

// ---------------------------------------------------------------- degree pass
__global__ __launch_bounds__(256) void k_init_deg(float* __restrict__ deg, int n) {
  int i = blockIdx.x * 256 + threadIdx.x;
  if (i < n) deg[i] = 1.0f;  // self-loop contributes 1 to in-degree
}

__global__ __launch_bounds__(256) void k_edge_deg(const int* __restrict__ dst,
                                                  float* __restrict__ deg, int e) {
  int i = blockIdx.x * 256 + threadIdx.x;
  if (i < e) {
    gatomic_fadd(&deg[dst[i]], 1.0f);
  }
}

__global__ __launch_bounds__(256) void k_rsqrt(float* __restrict__ dis, int n) {
  int i = blockIdx.x * 256 + threadIdx.x;
  if (i < n) dis[i] = rsqrtf(dis[i]);  // deg >= 1 always (self-loop)
}

// ---------------------------------------------------------------- zero agg
__global__ __launch_bounds__(256) void k_zero4(float4* __restrict__ p, int n4) {
  int i = blockIdx.x * 256 + threadIdx.x;
  if (i < n4) p[i] = make_float4(0.f, 0.f, 0.f, 0.f);
}

// ---------------------------------------------------------------- GEMM + row scale
// Hs[row] = (X @ W)[row] * dis[row]
// Block = 256 threads = 8 waves. Block owns 16 rows; wave w owns cols [16w,16w+16).
// K loop: 32 steps of v_wmma_f32_16x16x4_f32.
__global__ __launch_bounds__(256) void k_gemm_scale(const float* __restrict__ X,
                                                    const float* __restrict__ W,
                                                    const float* __restrict__ dis,
                                                    float* __restrict__ Hs, int n) {
  const int lane  = threadIdx.x & 31;
  const int wave  = threadIdx.x >> 5;
  const int l16   = lane & 15;
  const int h     = lane >> 4;            // lane-half selects K sub-pair
  const int baseM = blockIdx.x * 16;
  const int baseN = wave * 16;

  // clamped row pointer for A loads (tail-safe; wmma needs EXEC all-1s)
  int rowA = baseM + l16;
  if (rowA > n - 1) rowA = n - 1;
  const float* xrow = X + (size_t)rowA * FEAT;

  v8f c = {};
#pragma unroll
  for (int s = 0; s < 32; ++s) {
    const int k0 = 4 * s;
    // A 16x4 f32 fragment: VGPR0 = K{2h}, VGPR1 = K{2h+1}  (contiguous pair)
    v2f a = *(const v2f*)(xrow + k0 + 2 * h);
    // B 4x16 f32 fragment: rows K{2h}, K{2h+1} of W, column baseN+l16
    v2f b;
    b.x = W[(size_t)(k0 + 2 * h)     * FEAT + baseN + l16];
    b.y = W[(size_t)(k0 + 2 * h + 1) * FEAT + baseN + l16];
    c = __builtin_amdgcn_wmma_f32_16x16x4_f32(false, a, false, b,
                                              (short)0, c, false, false);
  }

  // C/D layout: VGPR r -> row (baseM + r + 8h), col (baseN + l16)
#pragma unroll
  for (int r = 0; r < 8; ++r) {
    int row = baseM + r + 8 * h;
    if (row < n) {
      Hs[(size_t)row * FEAT + baseN + l16] = c[r] * dis[row];
    }
  }
}

// ---------------------------------------------------------------- edge scatter
// 32 lanes per edge; each lane moves one float4 (512B row per edge, coalesced).
__global__ __launch_bounds__(256) void k_scatter(const int* __restrict__ src,
                                                 const int* __restrict__ dst,
                                                 const float* __restrict__ Hs,
                                                 float* __restrict__ agg, int e) {
  int gid  = blockIdx.x * 256 + threadIdx.x;
  int eidx = gid >> 5;
  int c4   = (gid & 31) << 2;  // feature offset 0..124 step 4
  if (eidx >= e) return;
  int s = src[eidx];
  int d = dst[eidx];
  const float4 v = *(const float4*)(Hs + (size_t)s * FEAT + c4);
  float* ap = agg + (size_t)d * FEAT + c4;
  gatomic_fadd(ap + 0, v.x);
  gatomic_fadd(ap + 1, v.y);
  gatomic_fadd(ap + 2, v.z);
  gatomic_fadd(ap + 3, v.w);
}

// ---------------------------------------------------------------- finalize
// x_out[v] = relu( dis[v] * (agg[v] + Hs[v]) + b )   (Hs[v] term = self-loop)
__global__ __launch_bounds__(256) void k_finalize(const float* __restrict__ Hs,
                                                  const float* __restrict__ agg,
                                                  const float* __restrict__ dis,
                                                  const float* __restrict__ b,
                                                  float* __restrict__ xout, int n) {
  int gid  = blockIdx.x * 256 + threadIdx.x;
  int node = gid >> 5;
  int c4   = (gid & 31) << 2;
  if (node >= n) return;
  float dv = dis[node];
  size_t idx = (size_t)node * FEAT + c4;
  float4 hv = *(const float4*)(Hs + idx);
  float4 av = *(const float4*)(agg + idx);
  float4 bv = *(const float4*)(b + c4);
  float4 o;
  o.x = fmaxf(dv * (av.x + hv.x) + bv.x, 0.0f);
  o.y = fmaxf(dv * (av.y + hv.y) + bv.y, 0.0f);
  o.z = fmaxf(dv * (av.z + hv.z) + bv.z, 0.0f);
  o.w = fmaxf(dv * (av.w + hv.w) + bv.w, 0.0f);
  *(float4*)(xout + idx) = o;
}

// ---------------------------------------------------------------- launch
extern "C" void kernel_launch(void* const* d_in, const int* in_sizes, int n_in,
                              void* d_out, int out_size, void* d_ws, size_t ws_size,
                              hipStream_t stream) {
  const float* x  = (const float*)d_in[0];
  const int*   ei = (const int*)d_in[1];
  const float* W  = (const float*)d_in[2];
  const float* b  = (const float*)d_in[3];

  const int n = in_sizes[0] / FEAT;
  const int e = in_sizes[1] / 2;
  const int* src = ei;
  const int* dst = ei + e;

  float* ws = (float*)d_ws;
  const size_t nf = (size_t)n * FEAT;
  float* dis = ws;
  size_t off = ((size_t)n + 255) & ~(size_t)255;
  float* Hs  = ws + off;
  float* agg = Hs + nf;
  float* xb  = agg + nf;

  const int nBlk  = (n + 255) / 256;
  const int eBlk  = (e + 255) / 256;
  const int zBlk  = (int)((nf / 4 + 255) / 256);
  const int gBlk  = (n + 15) / 16;
  const int sBlk  = (int)(((size_t)e * 32 + 255) / 256);
  const int fBlk  = (int)(((size_t)n * 32 + 255) / 256);

  // degree -> dis = deg^{-1/2}
  k_init_deg<<<nBlk, 256, 0, stream>>>(dis, n);
  k_edge_deg<<<eBlk, 256, 0, stream>>>(dst, dis, e);
  k_rsqrt<<<nBlk, 256, 0, stream>>>(dis, n);

  const float* xin = x;
  for (int t = 0; t < 3; ++t) {
    float* xout = (t == 2) ? (float*)d_out : xb;
    k_zero4<<<zBlk, 256, 0, stream>>>((float4*)agg, (int)(nf / 4));
    k_gemm_scale<<<gBlk, 256, 0, stream>>>(xin, W, dis, Hs, n);
    k_scatter<<<sBlk, 256, 0, stream>>>(src, dst, Hs, agg, e);
    k_finalize<<<fBlk, 256, 0, stream>>>(Hs, agg, dis, b, xout, n);
    xin = xb;
  }
}